// FLAMELearn_14336600834071
// MI455X (gfx1250) — compile-verified
//
#include <hip/hip_runtime.h>
#include <hip/hip_bf16.h>
#include <math.h>

// ---------------- problem constants ----------------
#define BB   1024
#define VV   5023
#define FF   9976
#define LLM  68
#define NSH  100
#define NEX  50
#define NJ   5
#define KBETA 152          // 150 padded to 38*4 for branchless WMMA K loop
#define KPOSE 36           // 9*4 exactly
#define RL   (3*VV)        // 15069 (posedirs row length)
#define OUTROW (VV + LLM)  // 5091 rows per batch in output
#define VTILES ((VV + 15) / 16)   // 314
#define BTILES (BB / 16)          // 64

typedef __attribute__((ext_vector_type(2))) float v2f;
typedef __attribute__((ext_vector_type(8))) float v8f;

// ---------------- kernel 0: fold J_regressor through shape space ----------------
// J_dirs[(j*3+c)*150 + l] = sum_v Jreg[j,v] * shapedirs[v,c,l]
// J_tmpl[j*3+c]           = sum_v Jreg[j,v] * v_template[v,c]
__global__ void fold_kernel(const float* __restrict__ Jreg,
                            const float* __restrict__ sdirs,
                            const float* __restrict__ vtmpl,
                            float* __restrict__ Jdirs,
                            float* __restrict__ Jtmpl) {
    int idx = blockIdx.x * blockDim.x + threadIdx.x;
    if (idx < 15 * 150) {
        int jc = idx / 150, l = idx % 150;
        int j = jc / 3, c = jc % 3;
        float s = 0.f;
        for (int v = 0; v < VV; ++v)
            s += Jreg[(size_t)j * VV + v] * sdirs[(size_t)(v * 3 + c) * 150 + l];
        Jdirs[idx] = s;
    } else if (idx < 15 * 150 + 15) {
        int jc = idx - 15 * 150;
        int j = jc / 3, c = jc % 3;
        float s = 0.f;
        for (int v = 0; v < VV; ++v)
            s += Jreg[(size_t)j * VV + v] * vtmpl[v * 3 + c];
        Jtmpl[jc] = s;
    }
}

// ---------------- rodrigues ----------------
__device__ __forceinline__ void rodrigues3(const float rv[3], float R[9]) {
    float a2 = rv[0]*rv[0] + rv[1]*rv[1] + rv[2]*rv[2] + 1e-16f;
    float ang = sqrtf(a2);
    float inv = 1.0f / ang;
    float x = rv[0]*inv, y = rv[1]*inv, z = rv[2]*inv;
    float s = sinf(ang), c = cosf(ang), t = 1.0f - c;
    R[0] = c + t*x*x;   R[1] = t*x*y - s*z; R[2] = t*x*z + s*y;
    R[3] = t*x*y + s*z; R[4] = c + t*y*y;   R[5] = t*y*z - s*x;
    R[6] = t*x*z - s*y; R[7] = t*y*z + s*x; R[8] = c + t*z*z;
}

// ---------------- kernel 1: per-batch prep ----------------
// one block per batch: betas (padded), J, rot_mats, pose_feature, A_rel
__global__ __launch_bounds__(160) void prep_kernel(
        const float* __restrict__ shape, const float* __restrict__ expr,
        const float* __restrict__ pose,  const float* __restrict__ eye,
        const float* __restrict__ Jdirs, const float* __restrict__ Jtmpl,
        float* __restrict__ betas, float* __restrict__ pf,
        float* __restrict__ Jws,   float* __restrict__ Arel) {
    const int b = blockIdx.x;
    const int t = threadIdx.x;

    if (t < KBETA) {
        float v = 0.f;
        if (t < NSH)            v = shape[(size_t)b * NSH + t];
        else if (t < NSH + NEX) v = expr[(size_t)b * NEX + (t - NSH)];
        betas[(size_t)b * KBETA + t] = v;
    }
    __syncthreads();

    if (t < 15) {  // J[b, j, c]
        float s = Jtmpl[t];
        const float* jd = Jdirs + t * 150;
        const float* be = betas + (size_t)b * KBETA;
        for (int l = 0; l < 150; ++l) s += jd[l] * be[l];
        Jws[b * 15 + t] = s;
    }
    __syncthreads();

    if (t == 0) {
        float fp[NJ][3];
        fp[0][0] = pose[b*6+0]; fp[0][1] = pose[b*6+1]; fp[0][2] = pose[b*6+2];
        fp[1][0] = 0.f; fp[1][1] = 0.f; fp[1][2] = 0.f;            // neck
        fp[2][0] = pose[b*6+3]; fp[2][1] = pose[b*6+4]; fp[2][2] = pose[b*6+5];
        fp[3][0] = eye[b*6+0];  fp[3][1] = eye[b*6+1];  fp[3][2] = eye[b*6+2];
        fp[4][0] = eye[b*6+3];  fp[4][1] = eye[b*6+4];  fp[4][2] = eye[b*6+5];

        float R[NJ][9];
        for (int j = 0; j < NJ; ++j) rodrigues3(fp[j], R[j]);

        // pose_feature = (rot_mats[1:] - I).flatten
        for (int j = 1; j < NJ; ++j)
            for (int r = 0; r < 3; ++r)
                for (int c = 0; c < 3; ++c)
                    pf[(size_t)b * KPOSE + (j-1)*9 + r*3 + c] =
                        R[j][r*3+c] - ((r == c) ? 1.f : 0.f);

        float Jv[NJ][3];
        for (int j = 0; j < NJ; ++j)
            for (int c = 0; c < 3; ++c) Jv[j][c] = Jws[b*15 + j*3 + c];

        const int par[NJ] = {-1, 0, 1, 1, 1};
        float rel[NJ][3];
        for (int c = 0; c < 3; ++c) rel[0][c] = Jv[0][c];
        for (int j = 1; j < NJ; ++j)
            for (int c = 0; c < 3; ++c) rel[j][c] = Jv[j][c] - Jv[par[j]][c];

        float Ra[NJ][9], ta[NJ][3];
        for (int q = 0; q < 9; ++q) Ra[0][q] = R[0][q];
        for (int c = 0; c < 3; ++c) ta[0][c] = rel[0][c];
        for (int j = 1; j < NJ; ++j) {
            int p = par[j];
            for (int r = 0; r < 3; ++r) {
                for (int c = 0; c < 3; ++c) {
                    float s = 0.f;
                    for (int k = 0; k < 3; ++k) s += Ra[p][r*3+k] * R[j][k*3+c];
                    Ra[j][r*3+c] = s;
                }
                ta[j][r] = Ra[p][r*3+0]*rel[j][0] + Ra[p][r*3+1]*rel[j][1]
                         + Ra[p][r*3+2]*rel[j][2] + ta[p][r];
            }
        }
        // A_rel: translation' = t - R @ J
        for (int j = 0; j < NJ; ++j) {
            for (int r = 0; r < 3; ++r) {
                float tr = ta[j][r] - (Ra[j][r*3+0]*Jv[j][0] +
                                       Ra[j][r*3+1]*Jv[j][1] +
                                       Ra[j][r*3+2]*Jv[j][2]);
                float* ar = Arel + (size_t)b*60 + j*12 + r*4;
                ar[0] = Ra[j][r*3+0]; ar[1] = Ra[j][r*3+1];
                ar[2] = Ra[j][r*3+2]; ar[3] = tr;
            }
        }
    }
}

// ---------------- kernel 2: fused shape+pose GEMMs + LBS skinning ----------------
// One wave per 16-batch x 16-vertex tile. M = batch, N = vertex, fp32 WMMA.
__global__ __launch_bounds__(256) void flame_main(
        const float* __restrict__ sdirs,   // (V,3,150)
        const float* __restrict__ pdirs,   // (36, V*3)
        const float* __restrict__ vtmpl,   // (V,3)
        const float* __restrict__ lbsw,    // (V,5)
        const float* __restrict__ trans,   // (B,3)
        const float* __restrict__ betas,   // (B,152) zero-padded
        const float* __restrict__ pf,      // (B,36)
        const float* __restrict__ Arel,    // (B,5,12)
        float* __restrict__ out) {         // (B, 5091, 3)
    const int lane = threadIdx.x & 31;
    const int wv   = threadIdx.x >> 5;
    const int tile = blockIdx.x * 8 + wv;
    const int vT = tile % VTILES;
    const int bT = tile / VTILES;

    const int ln = lane & 15;
    const int kg = lane >> 4;          // K-group: 0 or 1
    const int b_a = bT * 16 + ln;      // A-operand row (batch)
    const int v_n = vT * 16 + ln;      // B-operand / D column (vertex)
    const int v_c = (v_n < VV) ? v_n : (VV - 1);   // clamp for loads

    v8f acc0 = {}, acc1 = {}, acc2 = {};

    // ---- GEMM 1: v_shaped delta = betas @ shapedirs^T, K = 150 (padded 152) ----
    const float* ap   = betas + (size_t)b_a * KBETA;
    const float* brow = sdirs + (size_t)v_c * 450;   // 3 coords * 150
    for (int it = 0; it < 38; ++it) {
        const int kb  = it * 4 + kg * 2;
        const int kbc = (kb < 150) ? kb : 0;         // addr clamp; A is 0 on pad
        v2f av = *(const v2f*)(ap + kb);
        v2f b0 = *(const v2f*)(brow + kbc);
        v2f b1 = *(const v2f*)(brow + 150 + kbc);
        v2f b2 = *(const v2f*)(brow + 300 + kbc);
        acc0 = __builtin_amdgcn_wmma_f32_16x16x4_f32(false, av, false, b0, (short)0, acc0, false, false);
        acc1 = __builtin_amdgcn_wmma_f32_16x16x4_f32(false, av, false, b1, (short)0, acc1, false, false);
        acc2 = __builtin_amdgcn_wmma_f32_16x16x4_f32(false, av, false, b2, (short)0, acc2, false, false);
    }

    // ---- GEMM 2: pose blendshapes = pose_feature @ posedirs, K = 36 ----
    const float* ap2 = pf + (size_t)b_a * KPOSE;
    const int r0 = v_c * 3;
    for (int it = 0; it < 9; ++it) {
        const int kb = it * 4 + kg * 2;
        v2f av = *(const v2f*)(ap2 + kb);
        const float* p0 = pdirs + (size_t)kb * RL + r0;
        v2f b0 = { p0[0], p0[RL]     };
        v2f b1 = { p0[1], p0[RL + 1] };
        v2f b2 = { p0[2], p0[RL + 2] };
        acc0 = __builtin_amdgcn_wmma_f32_16x16x4_f32(false, av, false, b0, (short)0, acc0, false, false);
        acc1 = __builtin_amdgcn_wmma_f32_16x16x4_f32(false, av, false, b1, (short)0, acc1, false, false);
        acc2 = __builtin_amdgcn_wmma_f32_16x16x4_f32(false, av, false, b2, (short)0, acc2, false, false);
    }

    // ---- epilogue: + v_template, LBS blend of A_rel, + translation, store ----
    const int rowBase = kg * 8;                 // C/D rows held by this lane
    const float tp0 = vtmpl[v_c*3 + 0];
    const float tp1 = vtmpl[v_c*3 + 1];
    const float tp2 = vtmpl[v_c*3 + 2];
    const float* wp = lbsw + (size_t)v_c * NJ;
    const float w0 = wp[0], w1 = wp[1], w2 = wp[2], w3 = wp[3], w4 = wp[4];

    if (v_n < VV) {                              // divergence only after all WMMAs
        #pragma unroll
        for (int i = 0; i < 8; ++i) {
            const int b = bT * 16 + rowBase + i;
            const float* Ab = Arel + (size_t)b * 60;
            float T[12];
            #pragma unroll
            for (int q = 0; q < 12; ++q)
                T[q] = w0*Ab[q] + w1*Ab[12+q] + w2*Ab[24+q] + w3*Ab[36+q] + w4*Ab[48+q];
            const float vp0 = acc0[i] + tp0;
            const float vp1 = acc1[i] + tp1;
            const float vp2 = acc2[i] + tp2;
            const float o0 = T[0]*vp0 + T[1]*vp1 + T[2]*vp2  + T[3]  + trans[b*3+0];
            const float o1 = T[4]*vp0 + T[5]*vp1 + T[6]*vp2  + T[7]  + trans[b*3+1];
            const float o2 = T[8]*vp0 + T[9]*vp1 + T[10]*vp2 + T[11] + trans[b*3+2];
            float* op = out + ((size_t)b * OUTROW + v_n) * 3;
            op[0] = o0; op[1] = o1; op[2] = o2;
        }
    }
}

// ---------------- kernel 3: landmarks (barycentric gather from verts) ----------------
__global__ void lmk_kernel(const int* __restrict__ faces,
                           const int* __restrict__ lmkf,
                           const float* __restrict__ bary,
                           float* __restrict__ out) {
    int idx = blockIdx.x * blockDim.x + threadIdx.x;
    if (idx >= BB * LLM) return;
    int b = idx / LLM, l = idx % LLM;
    int f  = lmkf[l];
    int i0 = faces[f*3+0], i1 = faces[f*3+1], i2 = faces[f*3+2];
    float c0 = bary[l*3+0], c1 = bary[l*3+1], c2 = bary[l*3+2];
    const float* vb = out + (size_t)b * OUTROW * 3;
    float* op = out + ((size_t)b * OUTROW + VV + l) * 3;
    #pragma unroll
    for (int c = 0; c < 3; ++c)
        op[c] = c0 * vb[i0*3+c] + c1 * vb[i1*3+c] + c2 * vb[i2*3+c];
}

// ---------------- launch ----------------
extern "C" void kernel_launch(void* const* d_in, const int* in_sizes, int n_in,
                              void* d_out, int out_size, void* d_ws, size_t ws_size,
                              hipStream_t stream) {
    const float* shape = (const float*)d_in[0];
    const float* expr  = (const float*)d_in[1];
    const float* pose  = (const float*)d_in[2];
    const float* eye   = (const float*)d_in[3];
    const float* trans = (const float*)d_in[4];
    const float* vtmpl = (const float*)d_in[5];
    const float* sdirs = (const float*)d_in[6];
    const float* pdirs = (const float*)d_in[7];
    const float* Jreg  = (const float*)d_in[8];
    const float* lbsw  = (const float*)d_in[9];
    const int*   faces = (const int*)d_in[10];
    const int*   lmkf  = (const int*)d_in[11];
    const float* bary  = (const float*)d_in[12];
    float* out = (float*)d_out;

    float* ws    = (float*)d_ws;
    float* betas = ws;                       // B * 152
    float* pf    = betas + (size_t)BB * KBETA;   // B * 36
    float* Arel  = pf    + (size_t)BB * KPOSE;   // B * 60
    float* Jws   = Arel  + (size_t)BB * 60;      // B * 15
    float* Jdirs = Jws   + (size_t)BB * 15;      // 15 * 150
    float* Jtmpl = Jdirs + 15 * 150;             // 15

    fold_kernel<<<9, 256, 0, stream>>>(Jreg, sdirs, vtmpl, Jdirs, Jtmpl);
    prep_kernel<<<BB, 160, 0, stream>>>(shape, expr, pose, eye,
                                        Jdirs, Jtmpl, betas, pf, Jws, Arel);
    flame_main<<<(VTILES * BTILES) / 8, 256, 0, stream>>>(
        sdirs, pdirs, vtmpl, lbsw, trans, betas, pf, Arel, out);
    lmk_kernel<<<(BB * LLM + 255) / 256, 256, 0, stream>>>(faces, lmkf, bary, out);
}